// GraphEncoder_13211319403266
// MI455X (gfx1250) — compile-verified
//
#include <hip/hip_runtime.h>
#include <math.h>

typedef __attribute__((ext_vector_type(2))) float v2f;
typedef __attribute__((ext_vector_type(8))) float v8f;

#define NN    100000
#define EE    1600000
#define KK    50000
#define SORTN 131072  // next pow2 >= NN

// ---------- helpers ----------
__device__ __forceinline__ unsigned orderF(float f) {
  unsigned b = __float_as_uint(f);
  return (b & 0x80000000u) ? ~b : (b | 0x80000000u);  // monotone map: larger float -> larger uint
}

// ---------- C[M,128] = A[M,128] @ W[128,128]  via V_WMMA_F32_16X16X4_F32 ----------
// block = 256 threads = 8 wave32s; block computes 16 rows x 128 cols; wave w -> cols [16w,16w+16)
__global__ void k_gemm128(const float* __restrict__ A, const float* __restrict__ W,
                          float* __restrict__ C, int M) {
  __shared__ float As[16][132];  // +4 pad: column reads hit distinct banks
  const int row0 = blockIdx.x << 4;
  const int tid  = threadIdx.x;
  for (int i = tid; i < 16 * 128; i += 256) {
    int r = i >> 7, k = i & 127;
    int gr = row0 + r;
    As[r][k] = (gr < M) ? A[(size_t)gr * 128 + k] : 0.0f;
  }
  __syncthreads();
  const int wave = tid >> 5, lane = tid & 31;
  const int col0 = wave << 4;
  const int half = lane >> 4, t = lane & 15;
  v8f acc = {};
  for (int k0 = 0; k0 < 128; k0 += 4) {
    int ka = k0 + (half << 1);  // lanes 0-15: K=k0,k0+1 ; lanes 16-31: K=k0+2,k0+3
    v2f a, b;
    a.x = As[t][ka];
    a.y = As[t][ka + 1];
    b.x = W[(size_t)ka * 128 + col0 + t];
    b.y = W[(size_t)(ka + 1) * 128 + col0 + t];
    acc = __builtin_amdgcn_wmma_f32_16x16x4_f32(false, a, false, b, (short)0, acc, false, false);
  }
  // D layout: VGPR j holds row (j + 8*half), col t
  for (int j = 0; j < 8; ++j) {
    int r = row0 + j + (half << 3);
    if (r < M) C[(size_t)r * 128 + col0 + t] = acc[j];
  }
}

// ---------- degree counting / finalize ----------
__global__ void k_deg_count(const int* __restrict__ dst, unsigned* __restrict__ cnt, int e) {
  int i = blockIdx.x * blockDim.x + threadIdx.x;
  if (i < e) atomicAdd(&cnt[dst[i]], 1u);
}

__global__ void k_deg2_count(const int* __restrict__ src, const int* __restrict__ dst,
                             const int* __restrict__ newid, unsigned* __restrict__ cnt, int e) {
  int i = blockIdx.x * blockDim.x + threadIdx.x;
  if (i >= e) return;
  int s2 = newid[src[i]], d2 = newid[dst[i]];
  if (s2 >= 0 && d2 >= 0) atomicAdd(&cnt[d2], 1u);
}

__global__ void k_deg_fin(const unsigned* __restrict__ cnt, float* __restrict__ dinv,
                          float* __restrict__ invdeg, int n) {
  int i = blockIdx.x * blockDim.x + threadIdx.x;
  if (i >= n) return;
  float d = (float)(cnt[i] + 1u);  // +1 self-loop
  dinv[i] = rsqrtf(d);
  invdeg[i] = 1.0f / d;
}

// ---------- edge aggregation: wave per edge, 4 channels/lane, L2-resident atomics ----------
__global__ void k_edge_agg(const float* __restrict__ xw, const float* __restrict__ dinv,
                           const int* __restrict__ src, const int* __restrict__ dst,
                           float* __restrict__ agg, int e) {
  int g = blockIdx.x * blockDim.x + threadIdx.x;
  int wv = g >> 5, lane = g & 31;
  if (wv >= e) return;
  int s = src[wv], d = dst[wv];
  float w = dinv[s] * dinv[d];
  const float* xs = xw + (size_t)s * 128;
  float* ad = agg + (size_t)d * 128;
  for (int c = lane; c < 128; c += 32) unsafeAtomicAdd(&ad[c], xs[c] * w);
}

__global__ void k_edge_agg2(const float* __restrict__ xw, const float* __restrict__ dinv,
                            const int* __restrict__ newid, const int* __restrict__ src,
                            const int* __restrict__ dst, float* __restrict__ agg, int e) {
  int g = blockIdx.x * blockDim.x + threadIdx.x;
  int wv = g >> 5, lane = g & 31;
  if (wv >= e) return;
  int s2 = newid[src[wv]], d2 = newid[dst[wv]];
  if (s2 < 0 || d2 < 0) return;
  float w = dinv[s2] * dinv[d2];
  const float* xs = xw + (size_t)s2 * 128;
  float* ad = agg + (size_t)d2 * 128;
  for (int c = lane; c < 128; c += 32) unsafeAtomicAdd(&ad[c], xs[c] * w);
}

// ---------- out = relu(agg + xw*invdeg[node] + bias[c]) ----------
__global__ void k_combine(const float* __restrict__ agg, const float* __restrict__ xw,
                          const float* __restrict__ invdeg, const float* __restrict__ bias,
                          float* __restrict__ out, int n) {
  size_t i = (size_t)blockIdx.x * blockDim.x + threadIdx.x;
  if (i >= (size_t)n * 128) return;
  int node = (int)(i >> 7), c = (int)(i & 127);
  float v = agg[i] + xw[i] * invdeg[node] + bias[c];
  out[i] = v > 0.0f ? v : 0.0f;
}

// ---------- per-node scalar dots: hrel = h.Wrel, hroot = h.Wroot (wave per node) ----------
__global__ void k_dots(const float* __restrict__ h, const float* __restrict__ wrel,
                       const float* __restrict__ wroot, float* __restrict__ hrel,
                       float* __restrict__ hroot, int n) {
  int g = blockIdx.x * blockDim.x + threadIdx.x;
  int node = g >> 5, lane = g & 31;
  if (node >= n) return;
  const float* row = h + (size_t)node * 128;
  float s1 = 0.0f, s2 = 0.0f;
  for (int c = lane; c < 128; c += 32) {
    float v = row[c];
    s1 += v * wrel[c];
    s2 += v * wroot[c];
  }
  for (int m = 16; m >= 1; m >>= 1) {
    s1 += __shfl_xor(s1, m, 32);
    s2 += __shfl_xor(s2, m, 32);
  }
  if (lane == 0) { hrel[node] = s1; hroot[node] = s2; }
}

// segment_sum of hrel[src] into dst (linearized aggr@Wrel)
__global__ void k_sagg(const float* __restrict__ hrel, const int* __restrict__ src,
                       const int* __restrict__ dst, float* __restrict__ sagg, int e) {
  int i = blockIdx.x * blockDim.x + threadIdx.x;
  if (i < e) unsafeAtomicAdd(&sagg[dst[i]], hrel[src[i]]);
}

__global__ void k_score(const float* __restrict__ sagg, const float* __restrict__ hroot,
                        const float* __restrict__ brel, float* __restrict__ score, int n) {
  int i = blockIdx.x * blockDim.x + threadIdx.x;
  if (i < n) score[i] = tanhf(sagg[i] + brel[0] + hroot[i]);
}

// ---------- exact top-K: pack (ordered_score, ~index) -> u64, full bitonic descending sort ----------
__global__ void k_keys(const float* __restrict__ score, unsigned long long* __restrict__ keys, int n) {
  int i = blockIdx.x * blockDim.x + threadIdx.x;
  if (i >= SORTN) return;
  unsigned long long kv = 0ull;
  if (i < n)
    kv = ((unsigned long long)orderF(score[i]) << 32) |
         (unsigned long long)(0xFFFFFFFFu - (unsigned)i);  // ties: lower index wins in descending sort
  keys[i] = kv;
}

__global__ void k_bitonic(unsigned long long* __restrict__ keys, unsigned j, unsigned k) {
  unsigned i = blockIdx.x * blockDim.x + threadIdx.x;
  unsigned ixj = i ^ j;
  if (ixj <= i) return;
  unsigned long long a = keys[i], b = keys[ixj];
  bool desc = ((i & k) == 0);
  bool sw = desc ? (a < b) : (a > b);
  if (sw) { keys[i] = b; keys[ixj] = a; }
}

// gather kept nodes in rank order: hp[r] = h[idx]*score[idx]; newid[idx] = r
__global__ void k_pool(const unsigned long long* __restrict__ keys, const float* __restrict__ score,
                       const float* __restrict__ h, float* __restrict__ hp,
                       int* __restrict__ newid, int kkeep) {
  int gid = blockIdx.x * blockDim.x + threadIdx.x;
  int r = gid >> 7, c = gid & 127;
  if (r >= kkeep) return;
  unsigned long long kv = keys[r];
  int idx = (int)(0xFFFFFFFFu - (unsigned)(kv & 0xFFFFFFFFull));
  float sc = score[idx];
  hp[(size_t)r * 128 + c] = h[(size_t)idx * 128 + c] * sc;
  if (c == 0) newid[idx] = r;
}

// ---------- host orchestration ----------
extern "C" void kernel_launch(void* const* d_in, const int* in_sizes, int n_in,
                              void* d_out, int out_size, void* d_ws, size_t ws_size,
                              hipStream_t stream) {
  (void)in_sizes; (void)n_in; (void)out_size; (void)ws_size;
  const float* x     = (const float*)d_in[0];
  const int*   edge  = (const int*)d_in[1];
  const float* W1    = (const float*)d_in[2];
  const float* b1    = (const float*)d_in[3];
  const float* W2    = (const float*)d_in[4];
  const float* b2    = (const float*)d_in[5];
  const float* Wrel  = (const float*)d_in[6];
  const float* brel  = (const float*)d_in[7];
  const float* Wroot = (const float*)d_in[8];
  const int* src = edge;
  const int* dst = edge + EE;

  const size_t NB = (size_t)NN * 128;
  const size_t KB = (size_t)KK * 128;
  float* bufA    = (float*)d_ws;        // xw1 ; later hp (lower half) + xw2 (upper half)
  float* bufB    = bufA + NB;           // agg1 -> h ; later agg2 (lower KB)
  float* dinv1   = bufB + NB;
  float* invdeg1 = dinv1 + NN;
  float* hrel    = invdeg1 + NN;
  float* hroot   = hrel + NN;
  float* sagg    = hroot + NN;
  float* score   = sagg + NN;
  unsigned* cnt1 = (unsigned*)(score + NN);
  int* newid     = (int*)(cnt1 + NN);
  unsigned* cnt2 = (unsigned*)(newid + NN);
  float* dinv2   = (float*)(cnt2 + KK);
  float* invdeg2 = dinv2 + KK;
  unsigned long long* keys = (unsigned long long*)(invdeg2 + KK);

  // ---- conv1 ----
  hipMemsetAsync(bufB, 0, NB * sizeof(float), stream);
  hipMemsetAsync(cnt1, 0, NN * sizeof(unsigned), stream);
  hipMemsetAsync(sagg, 0, NN * sizeof(float), stream);
  k_gemm128<<<NN / 16, 256, 0, stream>>>(x, W1, bufA, NN);
  k_deg_count<<<(EE + 255) / 256, 256, 0, stream>>>(dst, cnt1, EE);
  k_deg_fin<<<(NN + 255) / 256, 256, 0, stream>>>(cnt1, dinv1, invdeg1, NN);
  k_edge_agg<<<EE / 8, 256, 0, stream>>>(bufA, dinv1, src, dst, bufB, EE);
  k_combine<<<(int)((NB + 255) / 256), 256, 0, stream>>>(bufB, bufA, invdeg1, b1, bufB, NN);  // h in-place

  // ---- SAGPool scoring ----
  k_dots<<<(NN + 7) / 8, 256, 0, stream>>>(bufB, Wrel, Wroot, hrel, hroot, NN);
  k_sagg<<<(EE + 255) / 256, 256, 0, stream>>>(hrel, src, dst, sagg, EE);
  k_score<<<(NN + 255) / 256, 256, 0, stream>>>(sagg, hroot, brel, score, NN);

  // ---- exact top-K via bitonic sort ----
  k_keys<<<SORTN / 256, 256, 0, stream>>>(score, keys, NN);
  for (unsigned kk = 2; kk <= SORTN; kk <<= 1)
    for (unsigned j = kk >> 1; j >= 1; j >>= 1)
      k_bitonic<<<SORTN / 256, 256, 0, stream>>>(keys, j, kk);
  hipMemsetAsync(newid, 0xFF, NN * sizeof(int), stream);  // -1 fill
  k_pool<<<(int)((KB + 255) / 256), 256, 0, stream>>>(keys, score, bufB, bufA, newid, KK);

  // ---- conv2 on pooled graph ----
  hipMemsetAsync(cnt2, 0, KK * sizeof(unsigned), stream);
  hipMemsetAsync(bufB, 0, KB * sizeof(float), stream);    // agg2 (h no longer needed)
  k_deg2_count<<<(EE + 255) / 256, 256, 0, stream>>>(src, dst, newid, cnt2, EE);
  k_deg_fin<<<(KK + 255) / 256, 256, 0, stream>>>(cnt2, dinv2, invdeg2, KK);
  k_gemm128<<<KK / 16, 256, 0, stream>>>(bufA, W2, bufA + KB, KK);
  k_edge_agg2<<<EE / 8, 256, 0, stream>>>(bufA + KB, dinv2, newid, src, dst, bufB, EE);
  k_combine<<<(int)((KB + 255) / 256), 256, 0, stream>>>(bufB, bufA + KB, invdeg2, b2,
                                                         (float*)d_out, KK);
}